// Capsule_91199335563673
// MI455X (gfx1250) — compile-verified
//
#include <hip/hip_runtime.h>
#include <stdint.h>

// Capsule dynamic routing, fused recompute strategy for MI455X (gfx1250).
// B=64 batch, J=2048 input capsules, I=16 in-dim, N=32 capsules, D=16 dim.
constexpr int CB  = 64;
constexpr int CJ  = 2048;
constexpr int CI  = 16;
constexpr int CN  = 32;
constexpr int CD  = 16;
constexpr int JWG = 8;                    // j's per workgroup
constexpr int BWG = 32;                   // b's per workgroup
constexpr int NJB = CJ / JWG;             // 256 j-blocks
constexpr int NWGS = NJB * (CB / BWG);    // 512 workgroups
constexpr int BP  = 48;                   // padded b-dim in A-pair layout (2*48%64==32)
constexpr int UROW = 20;                  // u_s row pitch (d-dim 16 + pad 4)
constexpr int UPLANE = BWG * UROW + 4;    // 644: plane pitch, 644%64==4 (bank spread)
constexpr float CEPS = 1e-7f;

typedef __attribute__((ext_vector_type(2))) float v2f;
typedef __attribute__((ext_vector_type(8))) float v8f;

// Workgroup: 32 waves (1024 thr), owns (b-half, 8-wide j block), all 32 capsules.
// Phase A (per j): wave w == capsule n computes u_hat[32b x 16d] via
//   2 btiles x 4 K-steps of v_wmma_f32_16x16x4_f32, stores tile to u_s.
// Phase B (per j): thread (b=wave, n=lane): 16-FMA logit dot, coalesced
//   b_state RMW, single wave-level softmax over n, 16-FMA c*u_hat accumulate.
__global__ __launch_bounds__(1024)
void routing_kernel(const float* __restrict__ x, const float* __restrict__ W,
                    float* __restrict__ bstate, const float* __restrict__ vprev,
                    float* __restrict__ spart)
{
    const int tid  = threadIdx.x;
    const int wave = tid >> 5;
    const int lane = tid & 31;
    const int half = lane >> 4;
    const int l16  = lane & 15;
    const int wg   = blockIdx.x;
    const int b0   = (wg & 1) * BWG;
    const int j0   = (wg >> 1) * JWG;

    __shared__ __align__(16) float x_s[JWG * 8 * BP * 2]; // A-fragment pairs
    __shared__ __align__(16) float u_s[CN * UPLANE];      // u_hat[n][b][d], padded

    // ---- async stage x straight into swizzled LDS (8B global <-> 8B LDS) ----
    {
        const uint32_t lds0 = (uint32_t)(uintptr_t)(&x_s[0]);
        #pragma unroll
        for (int rep = 0; rep < 2; ++rep) {
            const int idx = tid + rep * 1024;        // 0..2047
            const int p   = idx & 7;                 // pair index (i = 2p)
            const int jl  = (idx >> 3) & 7;
            const int b   = idx >> 6;                // 0..31
            const uint32_t goff = (uint32_t)((((b0 + b) * CJ + (j0 + jl)) * CI + 2 * p) * 4);
            const uint32_t loff = lds0 + (uint32_t)((((jl * 8 + p) * BP + b) * 2) * 4);
            asm volatile("global_load_async_to_lds_b64 %0, %1, %2"
                         :: "v"(loff), "v"(goff), "s"(x) : "memory");
        }
        asm volatile("s_wait_asynccnt 0" ::: "memory");
    }

    // Phase-B thread identity: b_local = wave, n = lane. Preload v_prev row.
    float vth[16];
    {
        const float* vp = vprev + ((size_t)(b0 + wave) * CN + lane) * CD;
        #pragma unroll
        for (int q = 0; q < 4; ++q) {
            const float4 t4 = *(const float4*)(vp + 4 * q);
            vth[4 * q + 0] = t4.x; vth[4 * q + 1] = t4.y;
            vth[4 * q + 2] = t4.z; vth[4 * q + 3] = t4.w;
        }
    }

    float sacc[16];
    #pragma unroll
    for (int d = 0; d < 16; ++d) sacc[d] = 0.f;

    __syncthreads();

    const float* Wn = W + (size_t)wave * CJ * (CD * CI);
    const v2f* xp = (const v2f*)x_s;             // pair view: [(jl*8+p)*BP + b]

    for (int jl = 0; jl < JWG; ++jl) {
        const int jg = j0 + jl;
        if (jl + 1 < JWG)
            __builtin_prefetch(Wn + (size_t)(jg + 1) * (CD * CI), 0, 1);

        // ---- Phase A: WMMA u_hat for capsule n = wave ----
        v2f bk[4];
        {
            const float* wj = Wn + (size_t)jg * (CD * CI);
            #pragma unroll
            for (int k = 0; k < 4; ++k)
                bk[k] = *(const v2f*)(wj + l16 * CI + k * 4 + half * 2);
        }

        #pragma unroll
        for (int t = 0; t < 2; ++t) {
            v8f acc = (v8f){};
            #pragma unroll
            for (int k = 0; k < 4; ++k) {
                const v2f a = xp[(jl * 8 + (k * 2 + half)) * BP + (t * 16 + l16)];
                acc = __builtin_amdgcn_wmma_f32_16x16x4_f32(
                        false, a, false, bk[k], (short)0, acc, false, false);
            }
            // store tile: lanes are d-consecutive (conflict-free b32 stores)
            float* ub = u_s + wave * UPLANE + (t * 16 + half * 8) * UROW + l16;
            #pragma unroll
            for (int r = 0; r < 8; ++r)
                ub[r * UROW] = acc[r];
        }

        __syncthreads();

        // ---- Phase B: routing for (b = wave, n = lane) ----
        float u[16];
        {
            const float* up = u_s + lane * UPLANE + wave * UROW;
            #pragma unroll
            for (int q = 0; q < 4; ++q) {
                const float4 t4 = *(const float4*)(up + 4 * q);
                u[4 * q + 0] = t4.x; u[4 * q + 1] = t4.y;
                u[4 * q + 2] = t4.z; u[4 * q + 3] = t4.w;
            }
        }

        float l = 0.f;
        #pragma unroll
        for (int d = 0; d < 16; ++d) l += u[d] * vth[d];

        // coalesced b_state RMW: layout [j][b][n] (private to this kernel)
        float* bsp = bstate + (size_t)jg * (CB * CN) + (b0 + wave) * CN + lane;
        const float lt = *bsp + l;
        *bsp = lt;

        // softmax over n == lane axis (single shuffle chain per wave)
        float m = lt;
        m = fmaxf(m, __shfl_xor(m, 1));
        m = fmaxf(m, __shfl_xor(m, 2));
        m = fmaxf(m, __shfl_xor(m, 4));
        m = fmaxf(m, __shfl_xor(m, 8));
        m = fmaxf(m, __shfl_xor(m, 16));
        const float e = __expf(lt - m);
        float s = e;
        s += __shfl_xor(s, 1);
        s += __shfl_xor(s, 2);
        s += __shfl_xor(s, 4);
        s += __shfl_xor(s, 8);
        s += __shfl_xor(s, 16);
        const float c = e / s;

        #pragma unroll
        for (int d = 0; d < 16; ++d) sacc[d] += c * u[d];

        __syncthreads();   // u_s reused next j
    }

    // partials: spart[wg][b_local][n][d], coalesced 16B-per-lane stores
    {
        float* sp = spart + (((size_t)wg * BWG + wave) * CN + lane) * CD;
        #pragma unroll
        for (int q = 0; q < 4; ++q) {
            float4 t4;
            t4.x = sacc[4 * q + 0]; t4.y = sacc[4 * q + 1];
            t4.z = sacc[4 * q + 2]; t4.w = sacc[4 * q + 3];
            *(float4*)(sp + 4 * q) = t4;
        }
    }
}

// Deterministic j-block reduction + squash. One thread per (b,n,d).
__global__ __launch_bounds__(256)
void reduce_squash_kernel(const float* __restrict__ spart, float* __restrict__ vout)
{
    const int gid = blockIdx.x * 256 + threadIdx.x;   // 0..32767
    const int d   = gid & 15;
    const int row = gid >> 4;                          // b*32 + n
    const int n   = row & 31;
    const int b   = row >> 5;
    const int bh  = b >> 5;                            // which b-half workgroup
    const int bl  = b & 31;

    float s = 0.f;
    for (int jb = 0; jb < NJB; ++jb)
        s += spart[(((size_t)(jb * 2 + bh) * BWG + bl) * CN + n) * CD + d];

    float s2 = s * s;                                  // norm over d: 16-lane group
    s2 += __shfl_xor(s2, 1);
    s2 += __shfl_xor(s2, 2);
    s2 += __shfl_xor(s2, 4);
    s2 += __shfl_xor(s2, 8);
    s2 += CEPS;
    const float scale = sqrtf(s2) / (1.0f + s2);
    vout[((size_t)b * CN + n) * CD + d] = scale * s;
}

extern "C" void kernel_launch(void* const* d_in, const int* in_sizes, int n_in,
                              void* d_out, int out_size, void* d_ws, size_t ws_size,
                              hipStream_t stream)
{
    const float* x = (const float*)d_in[0];   // [64,2048,16]
    const float* W = (const float*)d_in[1];   // [32,2048,16,16]
    float* out = (float*)d_out;               // [64,32,16]

    char* ws = (char*)d_ws;
    const size_t spart_bytes  = (size_t)NWGS * CN * BWG * CD * sizeof(float); // 32 MB
    const size_t bstate_bytes = (size_t)CB * CN * CJ * sizeof(float);         // 16 MB
    const size_t v_elems      = (size_t)CB * CN * CD;

    float* spart  = (float*)ws;
    float* bstate = (float*)(ws + spart_bytes);
    float* v0     = (float*)(ws + spart_bytes + bstate_bytes);
    float* v1     = v0 + v_elems;
    float* v2     = v1 + v_elems;

    // routing logits start at zero; v_prev = 0 reproduces uniform softmax iter 0
    hipMemsetAsync(bstate, 0, bstate_bytes, stream);
    hipMemsetAsync(v0, 0, v_elems * sizeof(float), stream);

    const dim3 gridR(NWGS), blockR(1024);
    const dim3 gridS((CB * CN * CD) / 256), blockS(256);

    routing_kernel<<<gridR, blockR, 0, stream>>>(x, W, bstate, v0, spart);
    reduce_squash_kernel<<<gridS, blockS, 0, stream>>>(spart, v1);

    routing_kernel<<<gridR, blockR, 0, stream>>>(x, W, bstate, v1, spart);
    reduce_squash_kernel<<<gridS, blockS, 0, stream>>>(spart, v2);

    routing_kernel<<<gridR, blockR, 0, stream>>>(x, W, bstate, v2, spart);
    reduce_squash_kernel<<<gridS, blockS, 0, stream>>>(spart, out);
}